// CategoryPrototype_60722247631620
// MI455X (gfx1250) — compile-verified
//
#include <hip/hip_runtime.h>

// Problem constants (match reference)
#define B_   4
#define N_   32768
#define D_   768
#define S_   1024
#define NMAD 5.0f
#define TPB  1024
#define KPT  (N_ / TPB)   // 32 points per thread

static __device__ __forceinline__ float neg_inf() { return -__builtin_inff(); }

// ---------------------------------------------------------------------------
// Block-wide sum reduction (all threads get the result). TPB threads.
// ---------------------------------------------------------------------------
static __device__ __forceinline__ unsigned blockSum(unsigned v, unsigned* sred) {
    const int t = threadIdx.x;
    sred[t] = v;
    __syncthreads();
    for (int off = TPB / 2; off > 0; off >>= 1) {
        if (t < off) sred[t] += sred[t + off];
        __syncthreads();
    }
    unsigned r = sred[0];
    __syncthreads();
    return r;
}

// ---------------------------------------------------------------------------
// Kernel 1: mask threshold + MAD depth-outlier rejection.
// One block per batch. Depth values cached in registers (32 per thread).
// k-th order statistic found by 32-step binary search over IEEE-754 bit
// patterns (all candidate values are >= 0 so the uint ordering is monotonic),
// counting via block reduction.
// ---------------------------------------------------------------------------
__global__ void __launch_bounds__(TPB)
mad_mask_kernel(const float* __restrict__ depth,
                const float* __restrict__ mask,
                unsigned char* __restrict__ wsmask) {
    __shared__ unsigned sred[TPB];
    const int b = blockIdx.x, t = threadIdx.x;
    const float* dep = depth + (size_t)b * N_;
    const float* msk = mask  + (size_t)b * N_;
    unsigned char* out = wsmask + (size_t)b * N_;

    float dv[KPT];
    unsigned vmbits = 0u;
    unsigned lc = 0u;
#pragma unroll
    for (int k = 0; k < KPT; ++k) {
        int i = k * TPB + t;
        dv[k] = dep[i];
        bool v = msk[i] >= 0.99f;
        vmbits |= (v ? 1u : 0u) << k;
        lc += v ? 1u : 0u;
    }
    const unsigned c = blockSum(lc, sred);   // uniform across block

    if (c == 0u) {           // no valid points: median undefined -> keep mask
#pragma unroll
        for (int k = 0; k < KPT; ++k)
            out[k * TPB + t] = (vmbits >> k) & 1u;
        return;
    }

    float med = 0.0f, mad = 0.0f;

    // countLE: number of masked values (depth or |depth-med|) with bits <= T
    auto countLE = [&](unsigned T, int pass) -> unsigned {
        unsigned cnt = 0u;
#pragma unroll
        for (int k = 0; k < KPT; ++k) {
            if ((vmbits >> k) & 1u) {
                float v = pass ? fabsf(dv[k] - med) : dv[k];
                if (__float_as_uint(v) <= T) cnt++;
            }
        }
        return blockSum(cnt, sred);
    };
    // k-th smallest (0-indexed) masked value; 32-iteration bisection on bits
    auto kth = [&](unsigned kk, int pass) -> float {
        unsigned lo = 0u, hi = 0xFFFFFFFFu;
        while (lo < hi) {                         // uniform loop (lo/hi uniform)
            unsigned mid = lo + ((hi - lo) >> 1);
            if (countLE(mid, pass) >= kk + 1u) hi = mid;
            else                               lo = mid + 1u;
        }
        return __uint_as_float(lo);
    };

    for (int pass = 0; pass < 2; ++pass) {        // 0: median, 1: MAD
        const unsigned k1 = (c - 1u) / 2u, k2 = c / 2u;
        float r1 = kth(k1, pass);
        float r2 = (k2 == k1) ? r1 : kth(k2, pass);   // uniform branch
        float m  = 0.5f * (r1 + r2);                  // nanmedian interpolation
        if (pass == 0) med = m; else mad = m;
    }

    // ret = m & (|d - med| < NMAD*mad); fallback to m if ret has no true entry
    const float thr = NMAD * mad;
    unsigned retbits = 0u, rc = 0u;
#pragma unroll
    for (int k = 0; k < KPT; ++k) {
        bool r = ((vmbits >> k) & 1u) && (fabsf(dv[k] - med) < thr);
        retbits |= (r ? 1u : 0u) << k;
        rc += r ? 1u : 0u;
    }
    const unsigned anyv = blockSum(rc, sred);
    const unsigned sel  = anyv ? retbits : vmbits;
#pragma unroll
    for (int k = 0; k < KPT; ++k)
        out[k * TPB + t] = (sel >> k) & 1u;
}

// ---------------------------------------------------------------------------
// Wave32 argmax reduce carrying (dist, idx, x, y, z); tie -> smaller index.
// ---------------------------------------------------------------------------
static __device__ __forceinline__ void waveArgmax(float& bd, int& bi,
                                                  float& bx, float& by, float& bz) {
    for (int off = 16; off > 0; off >>= 1) {
        float od = __shfl_xor(bd, off, 32);
        int   oi = __shfl_xor(bi, off, 32);
        float ox = __shfl_xor(bx, off, 32);
        float oy = __shfl_xor(by, off, 32);
        float oz = __shfl_xor(bz, off, 32);
        if (od > bd || (od == bd && oi < bi)) { bd = od; bi = oi; bx = ox; by = oy; bz = oz; }
    }
}

// ---------------------------------------------------------------------------
// Kernel 2: masked farthest-point sampling. One block (32 wave32 waves, fills
// a WGP) per batch. Points register-resident (96 f32/thread); the running
// min-squared-distance array lives in LDS (128 KB of the WGP's 320 KB).
// Squared distances preserve the reference's argmax/min ordering (sqrt is
// monotonic). Invalid points pinned at -inf.
// ---------------------------------------------------------------------------
__global__ void __launch_bounds__(TPB)
fps_kernel(const float* __restrict__ point_map,
           const unsigned char* __restrict__ wsmask,
           int* __restrict__ wsinds,
           float* __restrict__ outP) {
    __shared__ float sd[N_];          // 128 KB min-dist^2 state
    __shared__ int   sri[TPB];        // init-index reduce
    __shared__ float swd[32], swx[32], swy[32], swz[32];
    __shared__ int   swi[32];
    __shared__ float sq[3];           // broadcast of current farthest point

    const int b = blockIdx.x, t = threadIdx.x;
    const int lane = t & 31, w = t >> 5;
    const float* P = point_map + (size_t)b * N_ * 3;
    const unsigned char* M = wsmask + (size_t)b * N_;

    float px[KPT], py[KPT], pz[KPT];
    unsigned okbits = 0u;
#pragma unroll
    for (int k = 0; k < KPT; ++k) {
        int i = k * TPB + t;
        px[k] = P[(size_t)i * 3 + 0];
        py[k] = P[(size_t)i * 3 + 1];
        pz[k] = P[(size_t)i * 3 + 2];
        okbits |= (M[i] ? 1u : 0u) << k;
    }

    // initial sample = first valid index (0 if mask empty), per reference
    int mi = 0x7FFFFFFF;
#pragma unroll
    for (int k = 0; k < KPT; ++k) {
        int i = k * TPB + t;
        if (((okbits >> k) & 1u) && i < mi) mi = i;
    }
    sri[t] = mi;
    __syncthreads();
    for (int off = TPB / 2; off > 0; off >>= 1) {
        if (t < off) { int o = sri[t + off]; if (o < sri[t]) sri[t] = o; }
        __syncthreads();
    }
    const int idx0 = (sri[0] == 0x7FFFFFFF) ? 0 : sri[0];
    if (t == 0) {
        float qx = P[(size_t)idx0 * 3 + 0];
        float qy = P[(size_t)idx0 * 3 + 1];
        float qz = P[(size_t)idx0 * 3 + 2];
        sq[0] = qx; sq[1] = qy; sq[2] = qz;
        wsinds[b * S_ + 0] = idx0;
        size_t o = ((size_t)b * S_ + 0) * 3;
        outP[o] = qx; outP[o + 1] = qy; outP[o + 2] = qz;
    }
    __syncthreads();

    {   // initialize min-dist^2 against the initial point; invalid -> -inf
        float qx = sq[0], qy = sq[1], qz = sq[2];
#pragma unroll
        for (int k = 0; k < KPT; ++k) {
            int i = k * TPB + t;
            float dx = px[k] - qx, dy = py[k] - qy, dz = pz[k] - qz;
            float d2 = dx * dx + dy * dy + dz * dz;
            sd[i] = ((okbits >> k) & 1u) ? d2 : neg_inf();
        }
    }

    for (int s = 1; s < S_; ++s) {
        // per-thread argmax over its 32 slots (tie -> smaller index)
        float bd = neg_inf(); int bi = 0x7FFFFFFF;
        float bx = 0.f, by = 0.f, bz = 0.f;
#pragma unroll
        for (int k = 0; k < KPT; ++k) {
            int i = k * TPB + t;
            float v = sd[i];
            if (v > bd || (v == bd && i < bi)) { bd = v; bi = i; bx = px[k]; by = py[k]; bz = pz[k]; }
        }
        waveArgmax(bd, bi, bx, by, bz);
        if (lane == 0) { swd[w] = bd; swi[w] = bi; swx[w] = bx; swy[w] = by; swz[w] = bz; }
        __syncthreads();
        if (w == 0) {
            bd = swd[lane]; bi = swi[lane]; bx = swx[lane]; by = swy[lane]; bz = swz[lane];
            waveArgmax(bd, bi, bx, by, bz);
            if (lane == 0) {
                sq[0] = bx; sq[1] = by; sq[2] = bz;
                wsinds[b * S_ + s] = bi;
                size_t o = ((size_t)b * S_ + s) * 3;
                outP[o] = bx; outP[o + 1] = by; outP[o + 2] = bz;
            }
        }
        __syncthreads();
        // min-update against the newly selected point
        float qx = sq[0], qy = sq[1], qz = sq[2];
#pragma unroll
        for (int k = 0; k < KPT; ++k) {
            int i = k * TPB + t;
            float dx = px[k] - qx, dy = py[k] - qy, dz = pz[k] - qz;
            float nd = dx * dx + dy * dy + dz * dz;
            sd[i] = fminf(sd[i], nd);          // -inf (invalid) stays -inf
        }
    }
}

// ---------------------------------------------------------------------------
// Kernel 3: gather sampled feature rows via the CDNA5 async LDS DMA path.
// Each block copies one 768-f32 (3 KB) row: per-lane
// global_load_async_to_lds_b128 -> s_wait_asynccnt -> store_async_from_lds.
// Data never touches VGPRs; each wave drains only its own ASYNCcnt (waves own
// disjoint 512 B LDS slices), so no barrier is needed.
// ---------------------------------------------------------------------------
__global__ void __launch_bounds__(192)
gather_feats_kernel(const float* __restrict__ feats,
                    const int* __restrict__ wsinds,
                    float* __restrict__ outF) {
    __shared__ __align__(16) float stage[192 * 4];   // 3 KB staging
    const int bs = blockIdx.x;
    const int b  = bs / S_;
    const int r  = wsinds[bs];
    const float* src = feats + ((size_t)b * N_ + (size_t)r) * D_ + threadIdx.x * 4;
    float*       dst = outF + (size_t)bs * D_ + threadIdx.x * 4;
    unsigned lds = (unsigned)(unsigned long long)&stage[threadIdx.x * 4];

    asm volatile("global_load_async_to_lds_b128 %0, %1, off"
                 :: "v"(lds), "v"(src) : "memory");
    asm volatile("s_wait_asynccnt 0" ::: "memory");
    asm volatile("global_store_async_from_lds_b128 %0, %1, off"
                 :: "v"(dst), "v"(lds) : "memory");
    asm volatile("s_wait_asynccnt 0" ::: "memory");
}

// ---------------------------------------------------------------------------
extern "C" void kernel_launch(void* const* d_in, const int* in_sizes, int n_in,
                              void* d_out, int out_size, void* d_ws, size_t ws_size,
                              hipStream_t stream) {
    (void)in_sizes; (void)n_in; (void)out_size; (void)ws_size;
    const float* point_map = (const float*)d_in[0];   // (B,N,3)
    const float* depth     = (const float*)d_in[1];   // (B,N)
    const float* mask      = (const float*)d_in[2];   // (B,N)
    const float* feats     = (const float*)d_in[3];   // (B,N,D)

    float* outP = (float*)d_out;                       // (B,S,3)
    float* outF = outP + (size_t)B_ * S_ * 3;          // (B,S,D)

    unsigned char* wsmask = (unsigned char*)d_ws;                  // B*N bytes
    int* wsinds = (int*)(wsmask + (size_t)B_ * N_);                // B*S ints

    mad_mask_kernel<<<B_, TPB, 0, stream>>>(depth, mask, wsmask);
    fps_kernel<<<B_, TPB, 0, stream>>>(point_map, wsmask, wsinds, outP);
    gather_feats_kernel<<<B_ * S_, 192, 0, stream>>>(feats, wsinds, outF);
}